// Net_37160057045713
// MI455X (gfx1250) — compile-verified
//
#include <hip/hip_runtime.h>

// Five-level strided downsample (the reference conv stack is exactly
// x[:, :, ::2, ::2] applied 5 times), computed in a single pass over the
// even rows of the input.
//
// Traffic: ~403 MB read (even rows only) + ~268 MB written  ->  ~29 us at
// 23.3 TB/s HBM. Input rows are streamed global->LDS with gfx1250
// GLOBAL_LOAD_ASYNC_TO_LDS_B128 (ASYNCcnt-tracked, NT hint), pipelined with
// descending s_wait_asynccnt so compaction of chunk i overlaps the DMA of
// chunks i+1..3. Each thread consumes only the LDS bytes it loaded, so no
// workgroup barrier is required.

typedef __attribute__((ext_vector_type(2))) float v2f;

namespace {

constexpr int W  = 4096;   // input row length (floats)
constexpr int W1 = 2048;   // down_1 row length

// element offsets of each output inside the flat-concatenated d_out
constexpr unsigned long long O2 = 50331648ull;  // 12*2048*2048
constexpr unsigned long long O3 = 62914560ull;  // + 12*1024*1024
constexpr unsigned long long O4 = 66060288ull;  // + 12*512*512
constexpr unsigned long long O5 = 66846720ull;  // + 12*256*256

__global__ __launch_bounds__(256) void downsample5_kernel(
    const float* __restrict__ x, float* __restrict__ out)
{
    const int t  = threadIdx.x;   // 0..255
    const int r1 = blockIdx.x;    // down_1 row, 0..2047  -> input row 2*r1
    const int nc = blockIdx.y;    // fused N*C channel, 0..11

    __shared__ float lds[W];      // 16 KB: one full input row

    const float* src = x + (size_t)nc * W * W + (size_t)(2 * r1) * W;

    float* o1 = out      + (size_t)nc * (2048ull * 2048ull);
    float* o2 = out + O2 + (size_t)nc * (1024ull * 1024ull);
    float* o3 = out + O3 + (size_t)nc * (512ull * 512ull);
    float* o4 = out + O4 + (size_t)nc * (256ull * 256ull);
    float* o5 = out + O5 + (size_t)nc * (128ull * 128ull);

    // ---- Stage 1: issue 4 async B128 DMAs per thread (whole row -> LDS).
    // Thread t, chunk IT owns float4 #(IT*256 + t) of the row; identical
    // addressing is used in the consume stage, so no barrier is needed.
#define ISSUE(IT)                                                            \
    do {                                                                     \
        const int b_ = (IT) * 256 + t;                                       \
        const unsigned laddr_ =                                              \
            (unsigned)(unsigned long long)(&lds[4 * b_]);                    \
        const float* g_ = src + 4 * b_;                                      \
        asm volatile(                                                        \
            "global_load_async_to_lds_b128 %0, %1, off th:TH_LOAD_NT"        \
            :: "v"(laddr_), "v"(g_) : "memory");                             \
    } while (0)

    ISSUE(0); ISSUE(1); ISSUE(2); ISSUE(3);
#undef ISSUE

    // ---- Stage 2: pipelined consume. Async loads complete in order, so
    // s_wait_asynccnt (3-IT) guarantees chunks 0..IT have landed in LDS
    // while younger DMAs are still in flight.
#define STEP(IT, CNT)                                                        \
    do {                                                                     \
        asm volatile("s_wait_asynccnt " #CNT ::: "memory");                  \
        const int b_ = (IT) * 256 + t;      /* float4 index in row */        \
        const float e0_ = lds[4 * b_];      /* input col 4b   */             \
        const float e1_ = lds[4 * b_ + 2];  /* input col 4b+2 */             \
        v2f v2_; v2_.x = e0_; v2_.y = e1_;                                   \
        __builtin_nontemporal_store(                                         \
            v2_, (v2f*)(o1 + (size_t)r1 * W1 + 2 * b_));                     \
        if ((r1 & 1) == 0)                                                   \
            __builtin_nontemporal_store(                                     \
                e0_, o2 + (size_t)(r1 >> 1) * 1024 + b_);                    \
        if ((r1 & 3) == 0 && (b_ & 1) == 0)                                  \
            __builtin_nontemporal_store(                                     \
                e0_, o3 + (size_t)(r1 >> 2) * 512 + (b_ >> 1));              \
        if ((r1 & 7) == 0 && (b_ & 3) == 0)                                  \
            __builtin_nontemporal_store(                                     \
                e0_, o4 + (size_t)(r1 >> 3) * 256 + (b_ >> 2));              \
        if ((r1 & 15) == 0 && (b_ & 7) == 0)                                 \
            __builtin_nontemporal_store(                                     \
                e0_, o5 + (size_t)(r1 >> 4) * 128 + (b_ >> 3));              \
    } while (0)

    STEP(0, 3); STEP(1, 2); STEP(2, 1); STEP(3, 0);
#undef STEP
}

} // namespace

extern "C" void kernel_launch(void* const* d_in, const int* in_sizes, int n_in,
                              void* d_out, int out_size, void* d_ws, size_t ws_size,
                              hipStream_t stream)
{
    (void)in_sizes; (void)n_in; (void)out_size; (void)d_ws; (void)ws_size;

    const float* x = (const float*)d_in[0];   // (4,3,4096,4096) fp32
    float* out = (float*)d_out;               // 5 outputs concatenated flat

    // one block per (channel, down_1 row): 2048 x 12 = 24576 blocks
    dim3 grid(2048, 12, 1);
    dim3 block(256, 1, 1);
    downsample5_kernel<<<grid, block, 0, stream>>>(x, out);
}